// RoIRotate_29643864277589
// MI455X (gfx1250) — compile-verified
//
#include <hip/hip_runtime.h>

// RoIRotate (axis-aligned bilinear ROI sampling), MI455X / gfx1250.
//
// Shapes (fixed by the reference): B=8, J=64, C=32, HIMG=WIMG=256, HEIGHT=8.
// out = [res (B,J,C,8,max_w) f32][mask (B,J,max_w) f32], max_w from out_size.
//
// Memory-bound (~1 FLOP/byte). Design:
//  - 1 workgroup per box (512 WGs), 8 waves = 8 output rows, 32 lanes = w.
//  - Bilinear weights computed once per lane, reused across all 32 channels.
//  - Horizontal taps are adjacent -> one global_load_b64 per row per channel
//    (2 loads instead of 4; 4B-aligned b64 is legal in DWORD alignment mode).
//  - Non-temporal stores for the 105MB write-once output, keeping the 64MB
//    image resident in the 192MB L2 for the gathers.

#define ROI_HEIGHT 8
constexpr int kB = 8, kJ = 64, kC = 32, kHImg = 256, kWImg = 256;

__global__ __launch_bounds__(256) void roi_bilinear_kernel(
    const float* __restrict__ image,   // (B, C, 256, 256)
    const float* __restrict__ boxes,   // (B, J, 5) = left, top, right, bottom, theta
    float* __restrict__ out,           // (B, J, C, 8, max_w)
    float* __restrict__ mask,          // (B, J, max_w)
    int max_w)
{
  const int bj   = blockIdx.x;       // b*64 + j, 512 workgroups
  const int b    = bj >> 6;
  const int tid  = threadIdx.x;
  const int h    = tid >> 5;         // wave id == output row (8 waves/WG)
  const int lane = tid & 31;

  // Box parameters: wave-uniform (s_load_b128 path).
  const float left = boxes[bj * 5 + 0];
  const float top  = boxes[bj * 5 + 1];
  const float bw   = boxes[bj * 5 + 2] - left;
  const float bh   = boxes[bj * 5 + 3] - top;

  // Matches numpy: (bw / bh * 8).astype(int32) — f32 math, truncate.
  const int   Wbox   = (int)(bw / bh * (float)ROI_HEIGHT);
  const float each_w = bw / (float)(Wbox - 1);
  const float each_h = bh / (float)(ROI_HEIGHT - 1);

  // Vertical sample position: uniform per wave. Data guarantees
  // 0 <= top, top+bh <= 232, so clamps are no-ops kept only for safety.
  const float yf = fmaf((float)h, each_h, top);
  int y0 = (int)floorf(yf);
  int y1 = min(max(y0 + 1, 0), kHImg - 1);
  y0 = min(max(y0, 0), kHImg - 1);
  const float wy0 = (float)y1 - yf;   // weight for row y0  (y1f - y)
  const float wy1 = yf - (float)y0;   // weight for row y1  (y - y0f)

  const size_t plane = (size_t)kHImg * kWImg;            // 65536
  const float* img_b = image + (size_t)b * kC * plane;
  const float* row0  = img_b + (size_t)y0 * kWImg;       // channel 0, row y0
  const float* row1  = img_b + (size_t)y1 * kWImg;       // channel 0, row y1

  float* out_bh = out + ((size_t)bj * kC * ROI_HEIGHT + (size_t)h) * (size_t)max_w;
  const size_t cstride = (size_t)ROI_HEIGHT * (size_t)max_w;

  for (int wbase = 0; wbase < max_w; wbase += 32) {
    const int  w      = wbase + lane;
    const bool wvalid = (w < max_w);

    if (wbase < Wbox) {
      // Horizontal position: per lane, reused across all 32 channels.
      // In-box lanes: 0 <= x <= 240, so x0/x1 = floor(x)/floor(x)+1 with no
      // clamping (matches reference, whose clips never fire in-box).
      const float xf = fmaf((float)w, each_w, left);
      const int   x0 = (int)floorf(xf);
      float wx0 = (float)(x0 + 1) - xf;   // (x1f - x)
      float wx1 = xf - (float)x0;         // (x - x0f)
      const bool in = (w < Wbox);         // out-of-box lanes: zero weights,
      wx0 = in ? wx0 : 0.0f;              // address clamped -> no divergence
      wx1 = in ? wx1 : 0.0f;
      const float wa = wx0 * wy0;         // img[y0, x0]
      const float wb = wx0 * wy1;         // img[y1, x0]
      const float wc = wx1 * wy0;         // img[y0, x0+1]
      const float wd = wx1 * wy1;         // img[y1, x0+1]
      // Address-only clamp: keeps the 8-byte read inside the row window for
      // out-of-box lanes (whose weights are zero anyway).
      const size_t xa = (size_t)min(max(x0, 0), kWImg - 2);

      #pragma unroll 8
      for (int c = 0; c < kC; ++c) {
        const size_t cp = (size_t)c * plane;   // uniform -> saddr-form loads
        const float2 v0 = *reinterpret_cast<const float2*>(row0 + cp + xa);
        const float2 v1 = *reinterpret_cast<const float2*>(row1 + cp + xa);
        const float r = fmaf(v0.x, wa, fmaf(v1.x, wb, fmaf(v0.y, wc, v1.y * wd)));
        if (wvalid)  // streaming output: NT store, keep image in L2
          __builtin_nontemporal_store(r, out_bh + (size_t)c * cstride + (size_t)w);
      }
    } else {
      // Entire chunk past this box's width: wave-uniform zero fill.
      #pragma unroll 8
      for (int c = 0; c < kC; ++c)
        if (wvalid)
          __builtin_nontemporal_store(0.0f, out_bh + (size_t)c * cstride + (size_t)w);
    }

    if (h == 0 && wvalid)
      __builtin_nontemporal_store((w < Wbox) ? 1.0f : 0.0f,
                                  mask + (size_t)bj * (size_t)max_w + (size_t)w);
  }
}

extern "C" void kernel_launch(void* const* d_in, const int* in_sizes, int n_in,
                              void* d_out, int out_size, void* d_ws, size_t ws_size,
                              hipStream_t stream) {
  const float* image = (const float*)d_in[0];   // (8,32,256,256) f32
  const float* boxes = (const float*)d_in[1];   // (8,64,5)  f32
  float* out = (float*)d_out;

  const int BJ = kB * kJ;                                    // 512
  const int max_w = out_size / (BJ * (kC * ROI_HEIGHT + 1)); // 512*257*max_w
  float* mask = out + (size_t)BJ * kC * ROI_HEIGHT * (size_t)max_w;

  roi_bilinear_kernel<<<BJ, 256, 0, stream>>>(image, boxes, out, mask, max_w);
}